// SpikeFP32RoPE_48988396978590
// MI455X (gfx1250) — compile-verified
//
#include <hip/hip_runtime.h>
#include <cstdint>

// 16-byte clang ext-vector so __builtin_nontemporal_load/store lower to
// global_load_b128 / global_store_b128 with th:NT (HIP's uint4 is a struct,
// which the nontemporal builtins reject).
typedef uint32_t u32x4 __attribute__((ext_vector_type(4)));

#define HEAD_DIM  128
#define HALF      64
#define BITS      32
#define ROW_WORDS (HEAD_DIM * BITS)          // 4096 u32 per batch row
#define THETA_C   (-0.20762050593046016f)    // -log2(10000)/64

// One thread per (batch, d) with d in [0, 64): owns x1=x[b][d], x2=x[b][d+64].
// 512 B of global traffic per thread, all B128 + non-temporal (stream has zero
// reuse; don't rinse the 192 MB L2 with 512 MB of one-shot data).
__global__ __launch_bounds__(256) void spike_rope_stream(
    const uint32_t* __restrict__ x,      // pulse floats reinterpreted as u32
    const int*      __restrict__ position,
    uint32_t*       __restrict__ out,
    int total_pairs)                      // BATCH * 64
{
    int t = blockIdx.x * blockDim.x + threadIdx.x;
    if (t >= total_pairs) return;
    int b = t >> 6;
    int d = t & (HALF - 1);

    size_t row = (size_t)b * ROW_WORDS;
    const u32x4* __restrict__ p1 = (const u32x4*)(x + row + (size_t)d * BITS);
    const u32x4* __restrict__ p2 = (const u32x4*)(x + row + (size_t)(d + HALF) * BITS);

    u32x4 a[8], c[8];
#pragma unroll
    for (int j = 0; j < 8; ++j) a[j] = __builtin_nontemporal_load(p1 + j);
#pragma unroll
    for (int j = 0; j < 8; ++j) c[j] = __builtin_nontemporal_load(p2 + j);

    // Decode: pulse k (k=0 is MSB) -> bit (31-k). Pulses are exactly 0.0f
    // (0x00000000) or 1.0f (0x3F800000), so bit 23 of the raw word is the bit.
    uint32_t w1 = 0u, w2 = 0u;
#pragma unroll
    for (int j = 0; j < 8; ++j) {
        int k = 28 - 4 * j;  // j=0 fills bits 31..28
        uint32_t n1 = (((a[j].x >> 23) & 1u) << 3) | (((a[j].y >> 23) & 1u) << 2) |
                      (((a[j].z >> 23) & 1u) << 1) |  ((a[j].w >> 23) & 1u);
        uint32_t n2 = (((c[j].x >> 23) & 1u) << 3) | (((c[j].y >> 23) & 1u) << 2) |
                      (((c[j].z >> 23) & 1u) << 1) |  ((c[j].w >> 23) & 1u);
        w1 |= n1 << k;
        w2 |= n2 << k;
    }

    float xv1 = __uint_as_float(w1);
    float xv2 = __uint_as_float(w2);

    float pos   = (float)position[b];
    float theta = exp2f(THETA_C * (float)d);   // 10000^(-2d/128)
    float ang   = pos * theta;
    float sa, ca;
    sincosf(ang, &sa, &ca);

    float r1 = xv1 * ca - xv2 * sa;
    float r2 = xv2 * ca + xv1 * sa;
    uint32_t u1 = __float_as_uint(r1);
    uint32_t u2 = __float_as_uint(r2);

    // Encode: pulse k = bit (31-k) ? 1.0f : 0.0f, branchless:
    // (int)(w << k) >> 31 -> 0 or 0xFFFFFFFF, masked with 1.0f's bit pattern.
    u32x4* o1 = (u32x4*)(out + row + (size_t)d * BITS);
    u32x4* o2 = (u32x4*)(out + row + (size_t)(d + HALF) * BITS);
#pragma unroll
    for (int j = 0; j < 8; ++j) {
        int k = 4 * j;
        u32x4 v1, v2;
        v1.x = (uint32_t)((int32_t)(u1 << (k + 0)) >> 31) & 0x3F800000u;
        v1.y = (uint32_t)((int32_t)(u1 << (k + 1)) >> 31) & 0x3F800000u;
        v1.z = (uint32_t)((int32_t)(u1 << (k + 2)) >> 31) & 0x3F800000u;
        v1.w = (uint32_t)((int32_t)(u1 << (k + 3)) >> 31) & 0x3F800000u;
        v2.x = (uint32_t)((int32_t)(u2 << (k + 0)) >> 31) & 0x3F800000u;
        v2.y = (uint32_t)((int32_t)(u2 << (k + 1)) >> 31) & 0x3F800000u;
        v2.z = (uint32_t)((int32_t)(u2 << (k + 2)) >> 31) & 0x3F800000u;
        v2.w = (uint32_t)((int32_t)(u2 << (k + 3)) >> 31) & 0x3F800000u;
        __builtin_nontemporal_store(v1, o1 + j);
        __builtin_nontemporal_store(v2, o2 + j);
    }
}

// --- CDNA5 async-tensor path probe (compiled, never launched) ---------------
// Exercises GLOBAL_LOAD_ASYNC_TO_LDS_B128 + s_wait_asynccnt (CDNA5 ISA
// §4): VDST vgpr carries the LDS byte address, v[addr:addr+1] the 64-bit global
// address ("GV" mode, saddr=off). Confirms the mnemonic assembles on this
// toolchain in case we later pipeline the stream through LDS.
__global__ void cdna5_async_lds_probe(const uint32_t* __restrict__ g,
                                      uint32_t* __restrict__ o)
{
    __shared__ uint32_t lds[32 * 4];
    uint32_t lds_off = threadIdx.x * 16u;                 // bytes into LDS alloc
    uint64_t gaddr   = (uint64_t)g + (uint64_t)threadIdx.x * 16u;
    asm volatile("global_load_async_to_lds_b128 %0, %1, off"
                 :: "v"(lds_off), "v"(gaddr) : "memory");
    asm volatile("s_wait_asynccnt 0x0" ::: "memory");
    __syncthreads();
    o[threadIdx.x] = lds[threadIdx.x * 4];
}
// ---------------------------------------------------------------------------

extern "C" void kernel_launch(void* const* d_in, const int* in_sizes, int n_in,
                              void* d_out, int out_size, void* d_ws, size_t ws_size,
                              hipStream_t stream) {
    (void)n_in; (void)out_size; (void)d_ws; (void)ws_size;
    const uint32_t* x   = (const uint32_t*)d_in[0];  // fp32 pulses, raw bits
    const int*      pos = (const int*)d_in[1];
    uint32_t*       out = (uint32_t*)d_out;

    int batch       = in_sizes[0] / ROW_WORDS;       // 16384
    int total_pairs = batch * HALF;                  // 1,048,576 threads
    int block       = 256;
    int grid        = (total_pairs + block - 1) / block;

    spike_rope_stream<<<grid, block, 0, stream>>>(x, pos, out, total_pairs);
}